// AssociativeEmbeddingLoss_78812649882012
// MI455X (gfx1250) — compile-verified
//
#include <hip/hip_runtime.h>
#include <math.h>

// Problem constants (from reference): B=16, K=17, L=1, H=W=512, N=30
#define AE_B   16
#define AE_K   17
#define AE_N   30
#define AE_HW  (512 * 512)
#define AE_EPS 1e-6f
#define AE_LOSS_WEIGHT 1.0f
#define AE_PUSH_FACTOR 0.5f

typedef __attribute__((ext_vector_type(2))) float v2f;
typedef __attribute__((ext_vector_type(8))) float v8f;

// One wave32 per image. Lane n (< 30) owns person n.
__global__ __launch_bounds__(32) void ae_per_image_kernel(
    const float* __restrict__ tags,
    const int*   __restrict__ kpt_pos,
    const int*   __restrict__ kpt_vis,
    float*       __restrict__ partial)   // [2*B]: (pull_b, push_b)
{
    const int b    = blockIdx.x;
    const int lane = threadIdx.x;

    __shared__ float s_mean[32];
    __shared__ float s_valid[32];
    __shared__ float s_pull[32];
    __shared__ float s_push[32];

    float cnt   = 0.0f;
    float mean  = 0.0f;
    float pullv = 0.0f;

    if (lane < AE_N) {
        const int base = (b * AE_N + lane) * AE_K;
        float tk[AE_K];
        float vk[AE_K];
        float s = 0.0f;
#pragma unroll
        for (int k = 0; k < AE_K; ++k) {
            const int   p = kpt_pos[base + k];
            const float v = (float)kpt_vis[base + k];
            // channel index = k (L=1); gather one tag value
            const float t = tags[(size_t)(b * AE_K + k) * (size_t)AE_HW + (size_t)p];
            tk[k] = t;
            vk[k] = v;
            cnt += v;
            s   += v * t;
        }
        const float safe = fmaxf(cnt, 1.0f);
        mean = s / safe;                // 0 for fully-invisible persons (matches ref)
        float sq = 0.0f;
#pragma unroll
        for (int k = 0; k < AE_K; ++k) {
            const float d = tk[k] - mean;
            sq += d * d * vk[k];
        }
        pullv = sq / safe;              // L_ == 1
    }

    const float validf = (lane < AE_N && cnt > 0.0f) ? 1.0f : 0.0f;
    s_mean[lane]  = (lane < AE_N) ? mean : 0.0f;
    s_valid[lane] = validf;
    s_pull[lane]  = pullv;              // pull_per is 0 anyway when invalid
    __syncthreads();

    // ---------- Hoist ALL LDS reads into convergent register loads ----------
    // A/B operand means for this lane (column/row index = lane & 15):
    const int  lmod = lane & 15;
    const bool lo   = (lane < 16);
    float meanT[2], validT[2];
    meanT[0]  = s_mean[lmod];
    meanT[1]  = s_mean[16 + lmod];
    validT[0] = s_valid[lmod];
    validT[1] = s_valid[16 + lmod];

    // D-matrix row values this lane will need:
    //   lanes 0-15:  M = r      (VGPR r)      -> rowbase = 0
    //   lanes 16-31: M = 8 + r               -> rowbase = 8
    // mi_arr[ti*8 + r] = s_mean[ti*16 + rowbase + r]
    const int rowbase = (lane & 16) >> 1;    // 0 or 8, no divergence
    float mi_arr[16], vi_arr[16];
#pragma unroll
    for (int r = 0; r < 8; ++r) {
        mi_arr[r]     = s_mean [rowbase + r];
        vi_arr[r]     = s_valid[rowbase + r];
        mi_arr[8 + r] = s_mean [16 + rowbase + r];
        vi_arr[8 + r] = s_valid[16 + rowbase + r];
    }

    // ---- push term: pairwise products m_i * m_j via V_WMMA_F32_16X16X4_F32 ----
    // Outer product as a 16x4 * 4x16 f32 GEMM with only K=0 populated.
    // A layout (f32 16x4): lanes 0-15 hold row M=lane (VGPR0=K0, VGPR1=K1);
    //                      lanes 16-31 hold K2,K3 -> zeros there.
    float push_acc = 0.0f;
#pragma unroll
    for (int ti = 0; ti < 2; ++ti) {
#pragma unroll
        for (int tj = 0; tj < 2; ++tj) {
            v2f a;  a.x  = lo ? meanT[ti] : 0.0f;  a.y  = 0.0f;   // v_cndmask only
            v2f bm; bm.x = lo ? meanT[tj] : 0.0f;  bm.y = 0.0f;
            v8f c = {};
            c = __builtin_amdgcn_wmma_f32_16x16x4_f32(
                    /*neg_a=*/false, a, /*neg_b=*/false, bm,
                    /*c_mod=*/(short)0, c, /*reuse_a=*/false, /*reuse_b=*/false);

            const float mj = meanT[tj];
            const float vj = validT[tj];
#pragma unroll
            for (int r = 0; r < 8; ++r) {
                const float mi = mi_arr[ti * 8 + r];
                const float vi = vi_arr[ti * 8 + r];
                // (mi-mj)^2 = mi^2 + mj^2 - 2*mi*mj ; exactly 0 on diagonal
                const float d2 = mi * mi + mj * mj - 2.0f * c[r];
                push_acc += __expf(-d2) * (vi * vj);
            }
        }
    }

    s_push[lane] = push_acc;
    __syncthreads();

    if (lane == 0) {
        float Nv = 0.0f, pull_sum = 0.0f, push_sum = 0.0f;
        for (int i = 0; i < 32; ++i) {       // fixed order -> deterministic
            Nv       += s_valid[i];
            pull_sum += s_pull[i];
            push_sum += s_push[i];
        }
        const float pull = pull_sum / (Nv + AE_EPS);
        const float push = (Nv > 1.0f) ? (push_sum / ((Nv - 1.0f) * Nv + AE_EPS))
                                       : 0.0f;
        partial[2 * b + 0] = pull;
        partial[2 * b + 1] = push;
    }
}

// Deterministic fixed-order accumulation of the 16 per-image partials.
__global__ __launch_bounds__(32) void ae_finalize_kernel(
    const float* __restrict__ partial, float* __restrict__ out)
{
    if (threadIdx.x == 0 && blockIdx.x == 0) {
        float pull = 0.0f, push = 0.0f;
        for (int b = 0; b < AE_B; ++b) {
            pull += partial[2 * b + 0];
            push += partial[2 * b + 1];
        }
        out[0] = pull * AE_LOSS_WEIGHT;
        out[1] = push * AE_LOSS_WEIGHT * AE_PUSH_FACTOR;
    }
}

extern "C" void kernel_launch(void* const* d_in, const int* in_sizes, int n_in,
                              void* d_out, int out_size, void* d_ws, size_t ws_size,
                              hipStream_t stream)
{
    (void)in_sizes; (void)n_in; (void)out_size; (void)ws_size;

    const float* tags    = (const float*)d_in[0];
    const int*   kpt_pos = (const int*)d_in[1];
    const int*   kpt_vis = (const int*)d_in[2];
    float*       out     = (float*)d_out;
    float*       partial = (float*)d_ws;   // 32 floats of scratch

    ae_per_image_kernel<<<AE_B, 32, 0, stream>>>(tags, kpt_pos, kpt_vis, partial);
    ae_finalize_kernel<<<1, 32, 0, stream>>>(partial, out);
}